// LampSignature_33861522161705
// MI455X (gfx1250) — compile-verified
//
#include <hip/hip_runtime.h>
#include <hip/hip_bf16.h>

typedef float v2f __attribute__((ext_vector_type(2)));
typedef float v8f __attribute__((ext_vector_type(8)));

#define N_NODES 50000
#define IN_CH   256
#define HID     256
#define OUT_CH  128
#define N_EDGES 800000

// ---------------------------------------------------------------- degrees
__global__ void k_deg_init(float* __restrict__ deg) {
    int i = blockIdx.x * blockDim.x + threadIdx.x;
    if (i < N_NODES) deg[i] = 1.0f;              // self-loop contributes 1
}

__global__ void k_deg_edges(const int* __restrict__ ei, float* __restrict__ deg) {
    int e = blockIdx.x * blockDim.x + threadIdx.x;
    if (e < N_EDGES) atomicAdd(&deg[ei[e]], 1.0f);   // ei row 0 = src
}

__global__ void k_dinv(float* __restrict__ deg, float* __restrict__ svec) {
    int i = blockIdx.x * blockDim.x + threadIdx.x;
    if (i < N_NODES) {
        float d = deg[i];
        deg[i] = d > 0.0f ? rsqrtf(d) : 0.0f;    // dinv in place
    }
    if (i < HID) svec[i] = 0.0f;                 // zero the channel-sum vector
}

// ---------------------------------------------------------------- xw = x @ W (WMMA fp32)
// Block: 256 threads = 8 waves. Each wave: 16 rows x 64 cols. Block: 128 rows x 64 cols.
// B panel staged in LDS in pair-interleaved layout:
//   B[k][n]  ->  Bs[(k>>1)*128 + n*2 + (k&1)]
// so a lane's fragment (B[ka][n], B[ka+1][n]) is one aligned ds_load_b64.
__global__ __launch_bounds__(256) void k_gemm_xw(const float* __restrict__ x,
                                                 const float* __restrict__ w,
                                                 float* __restrict__ xw) {
    __shared__ float Bs[IN_CH * 64];             // 64 KB of the 320 KB WGP LDS
    const int col0 = blockIdx.y * 64;

    // cooperative fill: each thread grabs one (k2, n) pair -> two coalesced
    // global reads (consecutive n across lanes), one contiguous float2 LDS store
    for (int idx = threadIdx.x; idx < (IN_CH / 2) * 64; idx += 256) {
        int k2 = idx >> 6;                       // k pair index
        int n  = idx & 63;
        float a0 = w[(size_t)(2 * k2 + 0) * HID + col0 + n];
        float a1 = w[(size_t)(2 * k2 + 1) * HID + col0 + n];
        v2f p; p.x = a0; p.y = a1;
        *(v2f*)(Bs + (size_t)idx * 2) = p;       // Bs[k2*128 + n*2 + {0,1}]
    }
    __syncthreads();

    const int wave = threadIdx.x >> 5;
    const int lane = threadIdx.x & 31;
    const int row0 = blockIdx.x * 128 + wave * 16;
    if (row0 >= N_NODES) return;                 // wave-uniform; after barrier

    const int m    = lane & 15;                  // M index (A) / N index (B,D)
    const int half = lane >> 4;                  // K-half selector

    v8f acc[4];
    const v8f vzero = {0.f, 0.f, 0.f, 0.f, 0.f, 0.f, 0.f, 0.f};
    acc[0] = vzero; acc[1] = vzero; acc[2] = vzero; acc[3] = vzero;

    const float* xrow = x + (size_t)(row0 + m) * IN_CH;

    for (int kb = 0; kb < IN_CH; kb += 4) {
        const int ka = kb + half * 2;            // even by construction
        // A 16x4 fp32 fragment: lane holds A[m][ka], A[m][ka+1] (contiguous b64)
        v2f a = *(const v2f*)(xrow + ka);
        const float* brow = Bs + (ka >> 1) * 128;
#pragma unroll
        for (int t = 0; t < 4; t++) {
            const int n = t * 16 + m;
            // single ds_load_b64: (B[ka][n], B[ka+1][n])
            v2f b = *(const v2f*)(brow + n * 2);
            acc[t] = __builtin_amdgcn_wmma_f32_16x16x4_f32(
                false, a, false, b, (short)0, acc[t], false, false);
        }
    }

    // D layout: VGPR j, lanes0-15 -> M=j, lanes16-31 -> M=j+8; N = lane%16
#pragma unroll
    for (int j = 0; j < 8; j++) {
        const size_t r = (size_t)(row0 + j + half * 8);
        float* orow = xw + r * HID + col0;
        orow[ 0 + m] = acc[0][j];
        orow[16 + m] = acc[1][j];
        orow[32 + m] = acc[2][j];
        orow[48 + m] = acc[3][j];
    }
}

// ---------------------------------------------------------------- agg init with self-loop term
__global__ void k_selfloop(const float* __restrict__ xw, const float* __restrict__ dinv,
                           float* __restrict__ agg) {
    size_t idx = (size_t)blockIdx.x * blockDim.x + threadIdx.x;
    if (idx < (size_t)N_NODES * HID) {
        int i = (int)(idx >> 8);                 // /HID
        float di = dinv[i];
        agg[idx] = di * di * xw[idx];
    }
}

// ---------------------------------------------------------------- edge scatter-add (L2 atomics)
// One wave per edge; lane covers 8 channels (2 x float4 loads, 8 scalar f32 atomics).
__global__ __launch_bounds__(256) void k_edges(const int* __restrict__ ei,
                                               const float* __restrict__ dinv,
                                               const float* __restrict__ xw,
                                               float* __restrict__ agg) {
    int e = blockIdx.x * 8 + (threadIdx.x >> 5);
    if (e >= N_EDGES) return;
    int lane = threadIdx.x & 31;
    int s = ei[e];
    int d = ei[N_EDGES + e];
    float wgt = dinv[s] * dinv[d];

    const float4* xs = (const float4*)(xw + (size_t)s * HID);
    float* ad = agg + (size_t)d * HID;

    float4 v0 = xs[lane];                        // channels 4*lane .. +3
    float4 v1 = xs[lane + 32];                   // channels 128+4*lane .. +3
    int c0 = 4 * lane;
    atomicAdd(&ad[c0 + 0],       wgt * v0.x);
    atomicAdd(&ad[c0 + 1],       wgt * v0.y);
    atomicAdd(&ad[c0 + 2],       wgt * v0.z);
    atomicAdd(&ad[c0 + 3],       wgt * v0.w);
    atomicAdd(&ad[c0 + 128],     wgt * v1.x);
    atomicAdd(&ad[c0 + 129],     wgt * v1.y);
    atomicAdd(&ad[c0 + 130],     wgt * v1.z);
    atomicAdd(&ad[c0 + 131],     wgt * v1.w);
}

// ---------------------------------------------------------------- s = sum_n relu(agg + b)
__global__ __launch_bounds__(256) void k_reduce(const float* __restrict__ agg,
                                                const float* __restrict__ cb,
                                                float* __restrict__ svec) {
    int c = threadIdx.x;                         // channel
    float b = cb[c];
    float local = 0.0f;
    int n0 = blockIdx.x * 256;
    int n1 = n0 + 256; if (n1 > N_NODES) n1 = N_NODES;
    for (int n = n0; n < n1; n++) {
        float v = agg[(size_t)n * HID + c] + b;
        local += v > 0.0f ? v : 0.0f;
    }
    atomicAdd(&svec[c], local);
}

// ---------------------------------------------------------------- 4 GEMV heads + tanh
__global__ void k_heads(const float* __restrict__ svec,
                        const float* __restrict__ w1, const float* __restrict__ b1,
                        const float* __restrict__ w2, const float* __restrict__ b2,
                        const float* __restrict__ w3, const float* __restrict__ b3,
                        const float* __restrict__ w4, const float* __restrict__ b4,
                        float* __restrict__ out) {
    int o = blockIdx.x * blockDim.x + threadIdx.x;
    if (o >= 2 * HID + 2 * OUT_CH) return;       // 768 outputs
    const float* wr; float bb;
    if (o < 256)      { wr = w1 + (size_t)o * HID;         bb = b1[o]; }
    else if (o < 512) { wr = w2 + (size_t)(o - 256) * HID; bb = b2[o - 256]; }
    else if (o < 640) { wr = w3 + (size_t)(o - 512) * HID; bb = b3[o - 512]; }
    else              { wr = w4 + (size_t)(o - 640) * HID; bb = b4[o - 640]; }
    float acc = bb;
    for (int k = 0; k < HID; k++) acc += svec[k] * wr[k];
    out[o] = tanhf(acc);
}

// ----------------------------------------------------------------
extern "C" void kernel_launch(void* const* d_in, const int* in_sizes, int n_in,
                              void* d_out, int out_size, void* d_ws, size_t ws_size,
                              hipStream_t stream) {
    const float* x      = (const float*)d_in[0];
    const int*   ei     = (const int*)  d_in[1];
    const float* conv_w = (const float*)d_in[2];
    const float* conv_b = (const float*)d_in[3];
    const float* fc1_w  = (const float*)d_in[4];
    const float* fc1_b  = (const float*)d_in[5];
    const float* fc2_w  = (const float*)d_in[6];
    const float* fc2_b  = (const float*)d_in[7];
    const float* fc3_w  = (const float*)d_in[8];
    const float* fc3_b  = (const float*)d_in[9];
    const float* fc4_w  = (const float*)d_in[10];
    const float* fc4_b  = (const float*)d_in[11];
    float* out = (float*)d_out;

    // workspace layout (floats)
    float* ws   = (float*)d_ws;
    float* deg  = ws;                                     // 50000 (becomes dinv in place)
    float* svec = ws + 50176;                             // 256
    float* xw   = ws + 50432;                             // 50000*256
    float* agg  = xw + (size_t)N_NODES * HID;             // 50000*256

    // 1) degrees + dinv
    k_deg_init <<<(N_NODES + 255) / 256, 256, 0, stream>>>(deg);
    k_deg_edges<<<(N_EDGES + 255) / 256, 256, 0, stream>>>(ei, deg);
    k_dinv     <<<(N_NODES + 255) / 256, 256, 0, stream>>>(deg, svec);

    // 2) xw = x @ conv_w  (WMMA fp32, LDS-staged pair-interleaved B)
    dim3 ggrid((N_NODES + 127) / 128, HID / 64);
    k_gemm_xw<<<ggrid, 256, 0, stream>>>(x, conv_w, xw);

    // 3) agg = self-loop term, then edge scatter-add (L2-resident atomics)
    k_selfloop<<<(unsigned)(((size_t)N_NODES * HID + 255) / 256), 256, 0, stream>>>(xw, deg, agg);
    k_edges   <<<(N_EDGES + 7) / 8, 256, 0, stream>>>(ei, deg, xw, agg);

    // 4) s = sum_n relu(agg + conv_b)
    k_reduce<<<(N_NODES + 255) / 256, 256, 0, stream>>>(agg, conv_b, svec);

    // 5) heads
    k_heads<<<3, 256, 0, stream>>>(svec, fc1_w, fc1_b, fc2_w, fc2_b,
                                   fc3_w, fc3_b, fc4_w, fc4_b, out);
}